// EdgeFeature_11141145166318
// MI455X (gfx1250) — compile-verified
//
#include <hip/hip_runtime.h>
#include <hip/hip_bf16.h>
#include <stdint.h>

// EdgeFeature for point clouds: out[b,n,k,:] = concat(central(64), neighbor(64),
// neighbor-central(64), ||neighbor-central||^2 (1)).  B=4 N=8192 C=64 K=20.
// Bandwidth-bound: 506 MB output stream -> NT stores; point_cloud (8MB) stays
// hot in L2; central row staged to LDS via CDNA5 async-to-LDS DMA.

constexpr int Bb   = 4;
constexpr int Nn   = 8192;
constexpr int Cc   = 64;
constexpr int Kk   = 20;
constexpr int OUTC = 3 * Cc + 1;   // 193
constexpr int WAVES = 5;           // 5 waves/block
constexpr int KPW   = 4;           // 4 neighbors per wave -> 20 per block

__global__ __launch_bounds__(WAVES * 32)
void EdgeFeature_11141145166318_kernel(const float* __restrict__ pc,
                                       const int*   __restrict__ nn,
                                       float*       __restrict__ out)
{
    __shared__ float s_central[Cc];   // 256 B, offset 0 in block LDS

    const int bn   = blockIdx.x;          // one block per (b, n)
    const int b    = bn / Nn;
    const int n    = bn % Nn;
    const int lane = threadIdx.x & 31;
    const int wave = threadIdx.x >> 5;

    const float* crow = pc + ((size_t)b * Nn + n) * Cc;

    if (wave == 0) {
        // CDNA5 async DMA: 32 lanes x b64 = 256 B central row -> LDS.
        unsigned     lds_addr = (unsigned)(uintptr_t)(&s_central[0]) + (unsigned)lane * 8u;
        const float* src      = crow + lane * 2;
        asm volatile("global_load_async_to_lds_b64 %0, %1, off"
                     :: "v"(lds_addr), "v"(src) : "memory");
        asm volatile("s_wait_asynccnt 0x0" ::: "memory");
    }
    __syncthreads();

    const float c0 = s_central[lane];
    const float c1 = s_central[lane + 32];

    const int* krow = nn + ((size_t)b * Nn + n) * Kk;

    #pragma unroll
    for (int kk = 0; kk < KPW; ++kk) {
        const int k   = wave * KPW + kk;
        const int idx = krow[k];                       // wave-uniform -> scalar load
        const float* nrow = pc + ((size_t)b * Nn + idx) * Cc;

        if (kk + 1 < KPW) {
            // Warm L2/WGP$ for the next gathered row (global_prefetch_b8).
            const int idx2 = krow[k + 1];
            __builtin_prefetch(pc + ((size_t)b * Nn + idx2) * Cc, 0, 3);
        }

        const float n0 = nrow[lane];
        const float n1 = nrow[lane + 32];
        const float r0 = n0 - c0;
        const float r1 = n1 - c1;

        // Squared distance over C=64: 2 products/lane + wave32 XOR butterfly.
        float d = r0 * r0 + r1 * r1;
        #pragma unroll
        for (int off = 16; off >= 1; off >>= 1)
            d += __shfl_xor(d, off, 32);

        // 193-float output row; stream with non-temporal hint (output is 506 MB,
        // written once -> keep L2 for point_cloud reuse).
        float* orow = out + (((size_t)b * Nn + n) * Kk + k) * (size_t)OUTC;
        __builtin_nontemporal_store(c0, orow + lane);
        __builtin_nontemporal_store(c1, orow + 32 + lane);
        __builtin_nontemporal_store(n0, orow + 64 + lane);
        __builtin_nontemporal_store(n1, orow + 96 + lane);
        __builtin_nontemporal_store(r0, orow + 128 + lane);
        __builtin_nontemporal_store(r1, orow + 160 + lane);
        if (lane == 0)
            __builtin_nontemporal_store(d, orow + 192);
    }
}

extern "C" void kernel_launch(void* const* d_in, const int* in_sizes, int n_in,
                              void* d_out, int out_size, void* d_ws, size_t ws_size,
                              hipStream_t stream) {
    const float* pc  = (const float*)d_in[0];   // [B,N,C] f32
    const int*   nn  = (const int*)d_in[1];     // [B,N,K] i32
    float*       out = (float*)d_out;           // [B,N,K,3C+1] f32

    dim3 grid(Bb * Nn);        // 32768 blocks, one per (b,n)
    dim3 block(WAVES * 32);    // 160 threads = 5 wave32s
    hipLaunchKernelGGL(EdgeFeature_11141145166318_kernel, grid, block, 0, stream,
                       pc, nn, out);
}